// TLista_78975858639052
// MI455X (gfx1250) — compile-verified
//
#include <hip/hip_runtime.h>
#include <math.h>

// Problem constants (match reference)
#define NN 1024
#define MM 256
#define BB 2048
#define LL 10
#define KK (NN + MM)   // 1280: concatenated K dim of [W1 | W2] @ [x; y]

typedef float v2f __attribute__((ext_vector_type(2)));
typedef float v8f __attribute__((ext_vector_type(8)));

// ---------------------------------------------------------------------------
// Build per-layer A = [Toeplitz(v_l) | W2_l] as separate real / imag / -imag
// planes (f32 WMMA has no A-negate modifier, so we pre-negate for Cr term).
// ---------------------------------------------------------------------------
__global__ __launch_bounds__(256) void build_A_kernel(
    const float* __restrict__ vl,   // layer v: (2N-1) complex interleaved
    const float* __restrict__ W2l,  // layer W2: N*M complex interleaved
    float* __restrict__ Ar, float* __restrict__ Ai, float* __restrict__ Am)
{
    int idx = blockIdx.x * 256 + threadIdx.x;     // over N*K
    int i = idx / KK;
    int j = idx - i * KK;
    float re, im;
    if (j < NN) {
        int t = i - j + (NN - 1);                 // Toeplitz gather, v fits in cache
        re = vl[2 * t];
        im = vl[2 * t + 1];
    } else {
        int t = i * MM + (j - NN);
        re = W2l[2 * t];
        im = W2l[2 * t + 1];
    }
    Ar[idx] = re;
    Ai[idx] = im;
    Am[idx] = -im;
}

// ---------------------------------------------------------------------------
// Initialize RHS plane buffers: rows [0,N) of R0 = x0 = 0, rows [N,K) of both
// buffers = y (deinterleaved). R1 x-part is fully overwritten by layer 0.
// ---------------------------------------------------------------------------
__global__ __launch_bounds__(256) void init_R_kernel(
    const float* __restrict__ y,    // M*B complex interleaved
    float* __restrict__ R0r, float* __restrict__ R0i,
    float* __restrict__ R1r, float* __restrict__ R1i)
{
    int idx = blockIdx.x * 256 + threadIdx.x;     // over K*B
    int k = idx / BB;
    int col = idx - k * BB;
    if (k < NN) {
        R0r[idx] = 0.0f;
        R0i[idx] = 0.0f;
    } else {
        int t = (k - NN) * BB + col;
        float re = y[2 * t];
        float im = y[2 * t + 1];
        R0r[idx] = re; R0i[idx] = im;
        R1r[idx] = re; R1i[idx] = im;
    }
}

// ---------------------------------------------------------------------------
// One TLISTA layer: z = softthresh(A @ R, beta), complex via 4 real WMMAs.
// Wave tile: 32 rows x 64 cols (2 row strips share the 4 B column fragments,
// 4 col tiles share each A fragment). Block: 128 threads = 4 waves covering
// 128 rows x 64 cols.
// ---------------------------------------------------------------------------
__global__ __launch_bounds__(128) void tlista_layer_kernel(
    const float* __restrict__ Ar, const float* __restrict__ Ai,
    const float* __restrict__ Am,
    const float* __restrict__ Rr, const float* __restrict__ Ri,
    float* __restrict__ Or, float* __restrict__ Oi,   // plane outputs (non-final)
    float* __restrict__ OutC,                         // interleaved output (final)
    int finalLayer)
{
    const int lane = threadIdx.x & 31;
    const int wave = threadIdx.x >> 5;
    const int row0 = blockIdx.y * 128 + wave * 32;    // two 16-row strips
    const int col0 = blockIdx.x * 64;
    const int r  = lane & 15;
    const int hi = lane >> 4;   // 0: lanes 0-15, 1: lanes 16-31

    v8f cr[2][4] = {};   // real accumulators: [row strip][col tile]
    v8f ci[2][4] = {};   // imag accumulators

    // A row bases (A layout: VGPR0=K{0,2}, VGPR1=K{1,3} split by lane half)
    const float* arp0 = Ar + (size_t)(row0 + r) * KK;
    const float* aip0 = Ai + (size_t)(row0 + r) * KK;
    const float* amp0 = Am + (size_t)(row0 + r) * KK;
    const float* arp1 = arp0 + (size_t)16 * KK;
    const float* aip1 = aip0 + (size_t)16 * KK;
    const float* amp1 = amp0 + (size_t)16 * KK;

    for (int k0 = 0; k0 < KK; k0 += 4) {
        const int ka = k0 + 2 * hi;
        v2f fa_r0 = *(const v2f*)(arp0 + ka);
        v2f fa_i0 = *(const v2f*)(aip0 + ka);
        v2f fa_m0 = *(const v2f*)(amp0 + ka);
        v2f fa_r1 = *(const v2f*)(arp1 + ka);
        v2f fa_i1 = *(const v2f*)(aip1 + ka);
        v2f fa_m1 = *(const v2f*)(amp1 + ka);

        const float* brp = Rr + (size_t)ka * BB + col0 + r;
        const float* bip = Ri + (size_t)ka * BB + col0 + r;
#pragma unroll
        for (int t = 0; t < 4; ++t) {
            v2f fb_r, fb_i;
            fb_r.x = brp[t * 16];
            fb_r.y = brp[t * 16 + BB];
            fb_i.x = bip[t * 16];
            fb_i.y = bip[t * 16 + BB];
            // Row strip 0:  Cr += Ar*Br + (-Ai)*Bi ; Ci += Ar*Bi + Ai*Br
            cr[0][t] = __builtin_amdgcn_wmma_f32_16x16x4_f32(
                false, fa_r0, false, fb_r, (short)0, cr[0][t], false, false);
            cr[0][t] = __builtin_amdgcn_wmma_f32_16x16x4_f32(
                false, fa_m0, false, fb_i, (short)0, cr[0][t], false, false);
            ci[0][t] = __builtin_amdgcn_wmma_f32_16x16x4_f32(
                false, fa_r0, false, fb_i, (short)0, ci[0][t], false, false);
            ci[0][t] = __builtin_amdgcn_wmma_f32_16x16x4_f32(
                false, fa_i0, false, fb_r, (short)0, ci[0][t], false, false);
            // Row strip 1 (reuses the same B fragments)
            cr[1][t] = __builtin_amdgcn_wmma_f32_16x16x4_f32(
                false, fa_r1, false, fb_r, (short)0, cr[1][t], false, false);
            cr[1][t] = __builtin_amdgcn_wmma_f32_16x16x4_f32(
                false, fa_m1, false, fb_i, (short)0, cr[1][t], false, false);
            ci[1][t] = __builtin_amdgcn_wmma_f32_16x16x4_f32(
                false, fa_r1, false, fb_i, (short)0, ci[1][t], false, false);
            ci[1][t] = __builtin_amdgcn_wmma_f32_16x16x4_f32(
                false, fa_i1, false, fb_r, (short)0, ci[1][t], false, false);
        }
    }

    // Complex soft-threshold + store, using 1-ulp hardware sqrt/rcp.
    // C layout: VGPR e -> row (e + 8*hi), col = lane&15 within 16-wide tile.
#pragma unroll
    for (int s = 0; s < 2; ++s) {
#pragma unroll
        for (int t = 0; t < 4; ++t) {
#pragma unroll
            for (int e = 0; e < 8; ++e) {
                float zr = cr[s][t][e];
                float zi = ci[s][t][e];
                float m2  = zr * zr + zi * zi;
                float mag = __builtin_amdgcn_sqrtf(m2);          // v_sqrt_f32
                float sc  = fmaxf(mag - 0.01f, 0.0f) *
                            __builtin_amdgcn_rcpf(fmaxf(mag, 1e-12f)); // v_rcp_f32
                zr *= sc;
                zi *= sc;
                int row = row0 + s * 16 + e + 8 * hi;
                int col = col0 + t * 16 + r;
                size_t o = (size_t)row * BB + col;
                if (finalLayer) {
                    OutC[2 * o]     = zr;
                    OutC[2 * o + 1] = zi;
                } else {
                    Or[o] = zr;
                    Oi[o] = zi;
                }
            }
        }
    }
}

// ---------------------------------------------------------------------------
// Host-side launch: graph-capture safe (kernel launches on `stream` only).
// ws layout (floats):
//   Ar, Ai, Am        : N*K each          (3 * 1,310,720)
//   R0r, R0i, R1r, R1i: K*B each          (4 * 2,621,440)
// total ~57.7 MB.
// ---------------------------------------------------------------------------
extern "C" void kernel_launch(void* const* d_in, const int* in_sizes, int n_in,
                              void* d_out, int out_size, void* d_ws, size_t ws_size,
                              hipStream_t stream) {
    const float* v  = (const float*)d_in[0];   // (L, 2N-1) complex64 interleaved
    const float* W2 = (const float*)d_in[1];   // (L, N, M) complex64 interleaved
    const float* y  = (const float*)d_in[2];   // (M, B) complex64 interleaved
    float* out = (float*)d_out;                // (N, B) complex64 interleaved

    float* ws = (float*)d_ws;
    size_t off = 0;
    float* Ar  = ws + off; off += (size_t)NN * KK;
    float* Ai  = ws + off; off += (size_t)NN * KK;
    float* Am  = ws + off; off += (size_t)NN * KK;
    float* R0r = ws + off; off += (size_t)KK * BB;
    float* R0i = ws + off; off += (size_t)KK * BB;
    float* R1r = ws + off; off += (size_t)KK * BB;
    float* R1i = ws + off; off += (size_t)KK * BB;

    init_R_kernel<<<(KK * BB) / 256, 256, 0, stream>>>(y, R0r, R0i, R1r, R1i);

    float* curR_r = R0r; float* curR_i = R0i;
    float* nxtR_r = R1r; float* nxtR_i = R1i;

    for (int l = 0; l < LL; ++l) {
        const float* vl  = v  + (size_t)l * (2 * NN - 1) * 2;
        const float* W2l = W2 + (size_t)l * NN * MM * 2;
        build_A_kernel<<<(NN * KK) / 256, 256, 0, stream>>>(vl, W2l, Ar, Ai, Am);

        dim3 grid(BB / 64, NN / 128);   // (32, 8), 128-thread blocks
        int fin = (l == LL - 1) ? 1 : 0;
        tlista_layer_kernel<<<grid, 128, 0, stream>>>(
            Ar, Ai, Am, curR_r, curR_i, nxtR_r, nxtR_i, out, fin);

        // swap double buffers (y rows are identical in both)
        float* tr = curR_r; curR_r = nxtR_r; nxtR_r = tr;
        float* ti = curR_i; curR_i = nxtR_i; nxtR_i = ti;
    }
}